// HybridSort_17016660427269
// MI455X (gfx1250) — compile-verified
//
#include <hip/hip_runtime.h>
#include <hip/hip_bf16.h>

// NeuralSort hard forward: out[b,i,:] = one_hot(argmax_j (scaling_i * s_j - row_sum_j))
// B=8, N=2048. Output 128MiB f32 -> store-bandwidth bound.

#define kB 8
#define kN 2048

typedef __attribute__((ext_vector_type(2))) float v2f;
typedef __attribute__((ext_vector_type(8))) float v8f;

// ---------------------------------------------------------------------------
// Kernel 1: row_sum[b][i] = sum_k |s_i - s_k| via V_WMMA_F32_16X16X4_F32
// (the reference computes this as matmul(absdiff, ones*ones^T); we keep the
//  matmul formulation but only extract one column of D).
// grid = kB*8 blocks (batch b = blk>>3, i-chunk of 256 = blk&7), 256 threads.
// Each wave handles 32 rows i as two 16-row tiles; k is scanned 8 at a time
// with 4 independent accumulators (2 tiles x 2 k-phases) to break the
// WMMA->WMMA RAW chain.
// ---------------------------------------------------------------------------
__global__ __launch_bounds__(256) void rowsum_kernel(const float* __restrict__ scores,
                                                     float* __restrict__ rowsum) {
    __shared__ float s_lds[kN];
    const int b     = blockIdx.x >> 3;
    const int chunk = blockIdx.x & 7;
    const int tid   = threadIdx.x;

    // cooperative load of the whole batch row (8 KB) into LDS: 256 x 2 float4
    {
        const float4* s4 = (const float4*)(scores + b * kN);
        float4* l4 = (float4*)s_lds;
        l4[tid]       = s4[tid];
        l4[tid + 256] = s4[tid + 256];
    }
    __syncthreads();

    const int wave = tid >> 5;
    const int lane = tid & 31;
    const int half = lane >> 4;   // 0: lanes 0-15 (K=0,1), 1: lanes 16-31 (K=2,3)
    const int lm   = lane & 15;   // A-matrix row M within a 16-row tile

    const int i0 = chunk * 256 + wave * 32;  // tile 0: rows i0..i0+15
    const int i1 = i0 + 16;                  // tile 1: rows i1..i1+15

    const float si0 = s_lds[i0 + lm];
    const float si1 = s_lds[i1 + lm];

    v8f c00 = {}, c01 = {}, c10 = {}, c11 = {};
    v2f bones; bones[0] = 1.0f; bones[1] = 1.0f;  // B = 4x16 all-ones

    const float2* l2 = (const float2*)s_lds;
    for (int k0 = 0; k0 < kN; k0 += 8) {
        // phase A covers K = k0..k0+3, phase B covers K = k0+4..k0+7
        float2 va = l2[(k0 >> 1) + half];       // {s[k0+2*half], s[k0+2*half+1]}
        float2 vb = l2[(k0 >> 1) + half + 2];

        v2f a;
        a[0] = fabsf(si0 - va.x); a[1] = fabsf(si0 - va.y);
        c00 = __builtin_amdgcn_wmma_f32_16x16x4_f32(false, a, false, bones,
                                                    (short)0, c00, false, false);
        a[0] = fabsf(si0 - vb.x); a[1] = fabsf(si0 - vb.y);
        c01 = __builtin_amdgcn_wmma_f32_16x16x4_f32(false, a, false, bones,
                                                    (short)0, c01, false, false);
        a[0] = fabsf(si1 - va.x); a[1] = fabsf(si1 - va.y);
        c10 = __builtin_amdgcn_wmma_f32_16x16x4_f32(false, a, false, bones,
                                                    (short)0, c10, false, false);
        a[0] = fabsf(si1 - vb.x); a[1] = fabsf(si1 - vb.y);
        c11 = __builtin_amdgcn_wmma_f32_16x16x4_f32(false, a, false, bones,
                                                    (short)0, c11, false, false);
    }
    v8f c0 = c00 + c01;
    v8f c1 = c10 + c11;

    // D layout (16x16 f32): lanes 0-15 hold column=lane, M = vgpr;
    // lanes 16-31 hold column=lane-16, M = vgpr+8. All columns are identical
    // (B was all ones), so column 0 (lanes 0 and 16) has all 16 row sums.
    if (lm == 0) {
        float* dst = rowsum + b * kN;
        const int mb = half * 8;
        float4* o0 = (float4*)(dst + i0 + mb);
        float4* o1 = (float4*)(dst + i1 + mb);
        o0[0] = make_float4(c0[0], c0[1], c0[2], c0[3]);
        o0[1] = make_float4(c0[4], c0[5], c0[6], c0[7]);
        o1[0] = make_float4(c1[0], c1[1], c1[2], c1[3]);
        o1[1] = make_float4(c1[4], c1[5], c1[6], c1[7]);
    }
}

// ---------------------------------------------------------------------------
// Kernel 2: per (b,i) argmax_j of scal_i*s_j - row_sum_j, LDS-resident scan.
// Strict '>' with ascending j == jnp.argmax first-occurrence tie-break.
// grid = kB*16 blocks (i-chunks of 128), 128 threads, one i per thread.
// ---------------------------------------------------------------------------
__global__ __launch_bounds__(128) void argmax_kernel(const float* __restrict__ scores,
                                                     const float* __restrict__ rowsum,
                                                     int* __restrict__ idxbuf) {
    __shared__ float s_lds[kN];
    __shared__ float r_lds[kN];
    const int b     = blockIdx.x >> 4;
    const int chunk = blockIdx.x & 15;
    const int tid   = threadIdx.x;

    {
        const float4* s4 = (const float4*)(scores + b * kN);
        const float4* r4 = (const float4*)(rowsum + b * kN);
        float4* sl = (float4*)s_lds;
        float4* rl = (float4*)r_lds;
#pragma unroll
        for (int t = 0; t < 4; ++t) {
            sl[tid + t * 128] = s4[tid + t * 128];
            rl[tid + t * 128] = r4[tid + t * 128];
        }
    }
    __syncthreads();

    const int i = chunk * 128 + tid;
    const float scal = (float)(kN - 1 - 2 * i);

    float best = -3.4e38f;
    int   bidx = 0;
    const float4* sl4 = (const float4*)s_lds;
    const float4* rl4 = (const float4*)r_lds;
    for (int j4 = 0; j4 < kN / 4; ++j4) {
        float4 sv = sl4[j4];
        float4 rv = rl4[j4];
        float v0 = fmaf(scal, sv.x, -rv.x);
        float v1 = fmaf(scal, sv.y, -rv.y);
        float v2 = fmaf(scal, sv.z, -rv.z);
        float v3 = fmaf(scal, sv.w, -rv.w);
        const int j = j4 * 4;
        if (v0 > best) { best = v0; bidx = j;     }
        if (v1 > best) { best = v1; bidx = j + 1; }
        if (v2 > best) { best = v2; bidx = j + 2; }
        if (v3 > best) { best = v3; bidx = j + 3; }
    }
    idxbuf[b * kN + i] = bidx;
}

// ---------------------------------------------------------------------------
// Kernel 3: out[row, :] = one_hot(idx[row]).  128 MiB of wide streaming
// stores; one block per 8 KB row, each thread writes 8 floats as 2x b128.
// ---------------------------------------------------------------------------
__global__ __launch_bounds__(256) void onehot_kernel(const int* __restrict__ idxbuf,
                                                     float* __restrict__ out) {
    const int row = blockIdx.x;            // 0 .. kB*kN-1
    const int hot = idxbuf[row];           // wave-uniform scalar load
    const int j0  = threadIdx.x * 8;
    float* dst = out + (size_t)row * kN + j0;

    float4 z0, z1;
    z0.x = (hot == j0 + 0) ? 1.0f : 0.0f;
    z0.y = (hot == j0 + 1) ? 1.0f : 0.0f;
    z0.z = (hot == j0 + 2) ? 1.0f : 0.0f;
    z0.w = (hot == j0 + 3) ? 1.0f : 0.0f;
    z1.x = (hot == j0 + 4) ? 1.0f : 0.0f;
    z1.y = (hot == j0 + 5) ? 1.0f : 0.0f;
    z1.z = (hot == j0 + 6) ? 1.0f : 0.0f;
    z1.w = (hot == j0 + 7) ? 1.0f : 0.0f;

    ((float4*)dst)[0] = z0;
    ((float4*)dst)[1] = z1;
}

extern "C" void kernel_launch(void* const* d_in, const int* in_sizes, int n_in,
                              void* d_out, int out_size, void* d_ws, size_t ws_size,
                              hipStream_t stream) {
    const float* scores = (const float*)d_in[0];
    float* out = (float*)d_out;

    float* rowsum = (float*)d_ws;                                   // kB*kN f32 (64 KB)
    int*   idxbuf = (int*)((char*)d_ws + kB * kN * sizeof(float));  // kB*kN i32 (64 KB)

    rowsum_kernel<<<kB * 8, 256, 0, stream>>>(scores, rowsum);
    argmax_kernel<<<kB * 16, 128, 0, stream>>>(scores, rowsum, idxbuf);
    onehot_kernel<<<kB * kN, 256, 0, stream>>>(idxbuf, out);
}